// Attention_30382598652170
// MI455X (gfx1250) — compile-verified
//
#include <hip/hip_runtime.h>
#include <math.h>

typedef __attribute__((ext_vector_type(2))) float v2f;
typedef __attribute__((ext_vector_type(8))) float v8f;

#define DHc 128
#define DXc 128

// ---------------------------------------------------------------------------
// Kernel 1: proj = h @ a  via V_WMMA_F32_16X16X4_F32.
// One wave (32 lanes) computes one 16x16 tile of proj; K-loop of 32 WMMAs.
// A (16x4 f32) layout (ISA 7.12.2): lanes 0-15: M=lane, VGPR0=K0,VGPR1=K1;
//                                   lanes 16-31: M=lane-16, K2/K3.
// B (4x16 f32) symmetric: lanes carry N, halves carry K pairs.
// C/D (16x16 f32): VGPR v, lanes 0-15 -> M=v, lanes 16-31 -> M=v+8, N=lane&15.
// ---------------------------------------------------------------------------
__global__ __launch_bounds__(32) void proj_gemm_wmma(
    const float* __restrict__ h, const float* __restrict__ a,
    float* __restrict__ proj)
{
    const int lane  = threadIdx.x & 31;
    const int half  = lane >> 4;     // 0: K+0/K+1, 1: K+2/K+3
    const int l     = lane & 15;
    const int tileN = blockIdx.x;    // DX/16 tiles
    const int tileM = blockIdx.y;    // M/16 tiles

    const int rowA = tileM * 16 + l;
    const int colB = tileN * 16 + l;
    const float* hrow = h + (size_t)rowA * DHc;

    v8f acc = {};
    for (int k = 0; k < DHc; k += 4) {
        const int kk = k + half * 2;                       // even -> 8B aligned
        v2f av = *(const v2f*)(hrow + kk);                 // A: (rowA, kk..kk+1)
        v2f bv;
        bv.x = a[(size_t)kk * DXc + colB];                 // B: (kk,   colB)
        bv.y = a[(size_t)(kk + 1) * DXc + colB];           // B: (kk+1, colB)
        acc = __builtin_amdgcn_wmma_f32_16x16x4_f32(
            /*neg_a=*/false, av, /*neg_b=*/false, bv,
            /*c_mod=*/(short)0, acc, /*reuse_a=*/false, /*reuse_b=*/false);
    }

    const int mBase = tileM * 16 + half * 8;
    float* po = proj + (size_t)mBase * DXc + tileN * 16 + l;
#pragma unroll
    for (int v = 0; v < 8; ++v)
        po[(size_t)v * DXc] = acc[v];
}

// ---------------------------------------------------------------------------
// Kernel 2: fused segment softmax + attention pooling. One block per graph.
// segment_ids is sorted -> each graph owns a contiguous node range found by
// binary search. 4 waves/block, one node per wave per iteration; lane holds
// dims 4l..4l+3 (float4, fully coalesced b128 loads). Two passes over x
// (both L2-resident: x = 128MB < 192MB L2).
// ---------------------------------------------------------------------------
__device__ __forceinline__ int lbound(const int* __restrict__ s, int n, int v)
{
    int lo = 0, hi = n;
    while (lo < hi) { int mid = (lo + hi) >> 1; if (s[mid] < v) lo = mid + 1; else hi = mid; }
    return lo;
}

template <bool INLINE_PROJ>
__global__ __launch_bounds__(128) void attn_pool(
    const float* __restrict__ x, const int* __restrict__ seg,
    const float* __restrict__ proj, const float* __restrict__ h,
    const float* __restrict__ a, float* __restrict__ out, int N)
{
    const int g    = blockIdx.x;
    const int tid  = threadIdx.x;
    const int lane = tid & 31;
    const int wid  = tid >> 5;

    __shared__ __align__(16) float s_proj[DXc];
    __shared__ float  s_red[8];          // [0..3]=wave max, [4..7]=wave expsum
    __shared__ float4 s_part[4][32];     // per-wave output partials
    __shared__ int    s_range[2];

    if (tid == 0) {
        s_range[0] = lbound(seg, N, g);
        s_range[1] = lbound(seg, N, g + 1);
    }

    if (INLINE_PROJ) {
        // proj[g][tid] = sum_k h[g][k] * a[k][tid]  (h row is wave-uniform)
        float p = 0.f;
        const float* hrow = h + (size_t)g * DHc;
#pragma unroll 4
        for (int k = 0; k < DHc; ++k)
            p = fmaf(hrow[k], a[(size_t)k * DXc + tid], p);
        s_proj[tid] = p;
    } else {
        s_proj[tid] = proj[(size_t)g * DXc + tid];
    }
    __syncthreads();

    const int start = s_range[0], end = s_range[1];
    if (start >= end) {                          // empty segment -> zero row
        out[(size_t)g * DXc + tid] = 0.f;
        return;
    }

    const float4 pv = *(const float4*)(s_proj + 4 * lane);

    // ---- pass 1: per-wave online softmax stats over this wave's nodes ----
    float m = -INFINITY, z = 0.f;
    for (int n = start + wid; n < end; n += 4) {
        const float4 xv = *(const float4*)(x + (size_t)n * DXc + 4 * lane);
        __builtin_prefetch(x + (size_t)(n + 4) * DXc + 4 * lane, 0, 3);
        float d = xv.x * pv.x + xv.y * pv.y + xv.z * pv.z + xv.w * pv.w;
#pragma unroll
        for (int off = 16; off > 0; off >>= 1)
            d += __shfl_xor(d, off, 32);
        const float nm = fmaxf(m, d);
        z = z * __expf(m - nm) + __expf(d - nm);
        m = nm;
    }
    if (lane == 0) { s_red[wid] = m; s_red[4 + wid] = z; }
    __syncthreads();

    const float Mg = fmaxf(fmaxf(s_red[0], s_red[1]), fmaxf(s_red[2], s_red[3]));
    float Z = 0.f;
#pragma unroll
    for (int w = 0; w < 4; ++w)
        Z += s_red[4 + w] * __expf(s_red[w] - Mg);
    const float invZ = 1.0f / Z;

    // ---- pass 2: recompute score (x hits L2), accumulate att * x ----
    float4 acc = {0.f, 0.f, 0.f, 0.f};
    for (int n = start + wid; n < end; n += 4) {
        const float4 xv = *(const float4*)(x + (size_t)n * DXc + 4 * lane);
        __builtin_prefetch(x + (size_t)(n + 4) * DXc + 4 * lane, 0, 3);
        float d = xv.x * pv.x + xv.y * pv.y + xv.z * pv.z + xv.w * pv.w;
#pragma unroll
        for (int off = 16; off > 0; off >>= 1)
            d += __shfl_xor(d, off, 32);
        const float att = __expf(d - Mg) * invZ;
        acc.x = fmaf(att, xv.x, acc.x);
        acc.y = fmaf(att, xv.y, acc.y);
        acc.z = fmaf(att, xv.z, acc.z);
        acc.w = fmaf(att, xv.w, acc.w);
    }
    s_part[wid][lane] = acc;             // flat float idx = wid*128 + dim
    __syncthreads();

    const float* fp = (const float*)s_part;
    out[(size_t)g * DXc + tid] =
        fp[0 * DXc + tid] + fp[1 * DXc + tid] + fp[2 * DXc + tid] + fp[3 * DXc + tid];
}

// ---------------------------------------------------------------------------
// Host launcher
// inputs: h (M,128) f32 | x (N,128) f32 | a (128,128) f32 | segment_ids (N) i32
// output: (M,128) f32
// ---------------------------------------------------------------------------
extern "C" void kernel_launch(void* const* d_in, const int* in_sizes, int n_in,
                              void* d_out, int out_size, void* d_ws, size_t ws_size,
                              hipStream_t stream)
{
    const float* h   = (const float*)d_in[0];
    const float* x   = (const float*)d_in[1];
    const float* a   = (const float*)d_in[2];
    const int*   seg = (const int*)d_in[3];
    float* out = (float*)d_out;

    const int M = in_sizes[0] / DHc;   // 4096
    const int N = in_sizes[3];         // 262144

    float* proj = (float*)d_ws;
    const size_t projBytes = (size_t)M * DXc * sizeof(float);

    if (ws_size >= projBytes && (M % 16) == 0) {
        dim3 g1(DXc / 16, M / 16);     // (8, 256) tiles, 1 wave each
        proj_gemm_wmma<<<g1, 32, 0, stream>>>(h, a, proj);
        attn_pool<false><<<M, 128, 0, stream>>>(x, seg, proj, h, a, out, N);
    } else {
        attn_pool<true><<<M, 128, 0, stream>>>(x, seg, nullptr, h, a, out, N);
    }
}